// DeepFCNet_14946486190688
// MI455X (gfx1250) — compile-verified
//
#include <hip/hip_runtime.h>
#include <hip/hip_bf16.h>
#include <math.h>

// ---------------- problem constants ----------------
#define BB   32          // subjects
#define NN   264         // nets per subject
#define DD   375         // net feature dim
#define DDP  376         // padded K for layer 1
#define PP   34716       // NN*(NN-1)/2 pairs
#define ROWS (BB * NN)   // 8448 feature-extractor rows
#define SLOPE 0.2291666666666667f  // eval-mode RReLU slope 11/48

typedef float v2f __attribute__((ext_vector_type(2)));
typedef float v8f __attribute__((ext_vector_type(8)));

__device__ __forceinline__ v8f wmma_f32(v2f a, v2f b, v8f c) {
  return __builtin_amdgcn_wmma_f32_16x16x4_f32(false, a, false, b, (short)0, c,
                                               false, false);
}

__device__ __forceinline__ float rrelu(float t) {
  return (t >= 0.0f) ? t : t * SLOPE;
}

// ------- kernel 0: pair indices + zero c1 + padded weight copies ----------
__device__ __forceinline__ int pair_off(int i) {
  return (i * (2 * NN - 1 - i)) >> 1;   // start offset of row i in triu(k=1)
}

__global__ void prep_kernel(int* __restrict__ i1, int* __restrict__ i2,
                            float* __restrict__ c1,
                            const float* __restrict__ fe_w1,
                            float* __restrict__ fe_w1p,
                            const float* __restrict__ sm_w3,
                            float* __restrict__ sm_w3p) {
  int idx = blockIdx.x * blockDim.x + threadIdx.x;
  if (idx < BB * 4096) c1[idx] = 0.0f;
  if (idx < DDP * 64) fe_w1p[idx] = (idx < DD * 64) ? fe_w1[idx] : 0.0f;
  if (idx < 16 * 16) {
    int r = idx >> 4, c = idx & 15;
    sm_w3p[idx] = (c < 8) ? sm_w3[r * 8 + c] : 0.0f;
  }
  if (idx < PP) {
    float tn   = 2.0f * NN - 1.0f;
    float disc = tn * tn - 8.0f * (float)idx;
    int i = (int)((tn - sqrtf(fmaxf(disc, 0.0f))) * 0.5f);
    if (i < 0) i = 0;
    if (i > NN - 2) i = NN - 2;
    while (i > 0 && pair_off(i) > idx) --i;
    while (i < NN - 2 && pair_off(i + 1) <= idx) ++i;
    i1[idx] = i;
    i2[idx] = i + 1 + (idx - pair_off(i));
  }
}

// ---------------- kernel 1: feature extractor 375->64->64->32 -------------
__global__ __launch_bounds__(32) void fe_kernel(
    const float* __restrict__ x,
    const float* __restrict__ w1p, const float* __restrict__ b1,
    const float* __restrict__ w2, const float* __restrict__ b2,
    const float* __restrict__ w3, const float* __restrict__ b3,
    float* __restrict__ feats) {
  __shared__ float xt[16 * DDP];   // 16 rows, K padded 375 -> 376 (col 375 = 0)
  __shared__ float h1[16 * 64];
  __shared__ float h2[16 * 64];
  int lane = threadIdx.x;
  int row0 = blockIdx.x * 16;

  for (int idx = lane; idx < 16 * DDP; idx += 32) {
    int t = idx / DDP, c = idx - t * DDP;
    xt[idx] = (c < DD) ? x[(size_t)(row0 + t) * DD + c] : 0.0f;
  }
  __syncthreads();

  int lo = lane & 15;                 // A row == B col == C col
  int kh = (lane >> 4) << 1;          // K sub-offset per lane half
  int roff = (lane >> 4) * 8;         // C row base

  // layer 1: 375(->376)->64, rrelu
  for (int nt = 0; nt < 4; ++nt) {
    int nb = nt * 16;
    const float* ap = &xt[lo * DDP + kh];
    const float* wp = w1p + kh * 64 + nb + lo;
    v8f acc = {};
#pragma unroll 4
    for (int ks = 0; ks < DDP / 4; ++ks) {
      v2f a;  a.x = ap[0];  a.y = ap[1];
      v2f bf; bf.x = wp[0]; bf.y = wp[64];
      acc = wmma_f32(a, bf, acc);
      ap += 4; wp += 256;
    }
    float bias = b1[nb + lo];
    for (int v = 0; v < 8; ++v)
      h1[(roff + v) * 64 + nb + lo] = rrelu(acc[v] + bias);
  }
  __syncthreads();

  // layer 2: 64->64, relu
  for (int nt = 0; nt < 4; ++nt) {
    int nb = nt * 16;
    const float* ap = &h1[lo * 64 + kh];
    const float* wp = w2 + kh * 64 + nb + lo;
    v8f acc = {};
#pragma unroll 4
    for (int ks = 0; ks < 16; ++ks) {
      v2f a;  a.x = ap[0];  a.y = ap[1];
      v2f bf; bf.x = wp[0]; bf.y = wp[64];
      acc = wmma_f32(a, bf, acc);
      ap += 4; wp += 256;
    }
    float bias = b2[nb + lo];
    for (int v = 0; v < 8; ++v)
      h2[(roff + v) * 64 + nb + lo] = fmaxf(acc[v] + bias, 0.0f);
  }
  __syncthreads();

  // layer 3: 64->32, linear
  for (int nt = 0; nt < 2; ++nt) {
    int nb = nt * 16;
    const float* ap = &h2[lo * 64 + kh];
    const float* wp = w3 + kh * 32 + nb + lo;
    v8f acc = {};
#pragma unroll 4
    for (int ks = 0; ks < 16; ++ks) {
      v2f a;  a.x = ap[0];  a.y = ap[1];
      v2f bf; bf.x = wp[0]; bf.y = wp[32];
      acc = wmma_f32(a, bf, acc);
      ap += 4; wp += 128;
    }
    float bias = b3[nb + lo];
    for (int v = 0; v < 8; ++v)
      feats[(size_t)(row0 + roff + v) * 32 + nb + lo] = acc[v] + bias;
  }
}

// ---------------- kernel 2: similarity MLP over pairs ----------------
__global__ __launch_bounds__(32) void sim_kernel(
    const float* __restrict__ feats,
    const int* __restrict__ i1, const int* __restrict__ i2,
    const float* __restrict__ w1, const float* __restrict__ b1,
    const float* __restrict__ w2, const float* __restrict__ b2,
    const float* __restrict__ w3p, const float* __restrict__ b3,
    const float* __restrict__ w4, const float* __restrict__ b4,
    float* __restrict__ s) {
  __shared__ float pt[16 * 64];
  __shared__ float h1[16 * 32];
  __shared__ float h2[16 * 16];
  __shared__ float h3[16 * 8];
  __shared__ int ia[16], ib[16];

  int lane = threadIdx.x;
  int b    = blockIdx.y;
  int p0   = blockIdx.x * 16;

  if (lane < 16) {
    int p = p0 + lane;
    if (p >= PP) p = PP - 1;      // duplicate last pair for padded rows
    ia[lane] = i1[p];
    ib[lane] = i2[p];
  }
  __syncthreads();

  const float* fb = feats + (size_t)b * NN * 32;
  for (int idx = lane; idx < 16 * 64; idx += 32) {
    int t = idx >> 6, c = idx & 63;
    pt[idx] = (c < 32) ? fb[ia[t] * 32 + c] : fb[ib[t] * 32 + (c - 32)];
  }
  __syncthreads();

  int lo = lane & 15, kh = (lane >> 4) << 1, roff = (lane >> 4) * 8;

  // L1: 64->32, relu
  for (int nt = 0; nt < 2; ++nt) {
    int nb = nt * 16;
    const float* ap = &pt[lo * 64 + kh];
    const float* wp = w1 + kh * 32 + nb + lo;
    v8f acc = {};
#pragma unroll 4
    for (int ks = 0; ks < 16; ++ks) {
      v2f a;  a.x = ap[0];  a.y = ap[1];
      v2f bf; bf.x = wp[0]; bf.y = wp[32];
      acc = wmma_f32(a, bf, acc);
      ap += 4; wp += 128;
    }
    float bias = b1[nb + lo];
    for (int v = 0; v < 8; ++v)
      h1[(roff + v) * 32 + nb + lo] = fmaxf(acc[v] + bias, 0.0f);
  }
  __syncthreads();

  // L2: 32->16, relu
  {
    const float* ap = &h1[lo * 32 + kh];
    const float* wp = w2 + kh * 16 + lo;
    v8f acc = {};
#pragma unroll 4
    for (int ks = 0; ks < 8; ++ks) {
      v2f a;  a.x = ap[0];  a.y = ap[1];
      v2f bf; bf.x = wp[0]; bf.y = wp[16];
      acc = wmma_f32(a, bf, acc);
      ap += 4; wp += 64;
    }
    float bias = b2[lo];
    for (int v = 0; v < 8; ++v)
      h2[(roff + v) * 16 + lo] = fmaxf(acc[v] + bias, 0.0f);
  }
  __syncthreads();

  // L3: 16->8 via padded [16][16] weights (cols 8..15 zero), relu
  {
    const float* ap = &h2[lo * 16 + kh];
    const float* wp = w3p + kh * 16 + lo;
    v8f acc = {};
#pragma unroll
    for (int ks = 0; ks < 4; ++ks) {
      v2f a;  a.x = ap[0];  a.y = ap[1];
      v2f bf; bf.x = wp[0]; bf.y = wp[16];
      acc = wmma_f32(a, bf, acc);
      ap += 4; wp += 64;
    }
    if (lo < 8) {
      float bias = b3[lo];
      for (int v = 0; v < 8; ++v)
        h3[(roff + v) * 8 + lo] = fmaxf(acc[v] + bias, 0.0f);
    }
  }
  __syncthreads();

  // L4: 8->1 + tanh, one pair per lane 0..15
  if (lane < 16) {
    float acc = b4[0];
    for (int k = 0; k < 8; ++k) acc += h3[lane * 8 + k] * w4[k];
    int p = p0 + lane;
    if (p < PP) s[(size_t)b * PP + p] = tanhf(acc);
  }
}

// ------- kernel 3: big GEMM  c1 += s[32,P] @ cl_w1[P,4096]  (split-K) ------
// 32 output cols x 32 rows per wave. Two-phase blocked loop: a clause of 48
// loads (8 k-steps) is issued before the 32 WMMAs consume them, keeping ~6KB
// per wave in flight (4224 waves -> ~25MB outstanding > HBM BDP ~14MB).
#define KCHUNKS 33
#define KPER    263    // 33 * 263 * 4 == 34716 exactly
#define TBLK    8      // k-steps per load/compute block (263 = 32*8 + 7)
__global__ __launch_bounds__(32) void cls1_kernel(
    const float* __restrict__ s, const float* __restrict__ w1,
    float* __restrict__ c1) {
  int lane = threadIdx.x;
  int nb   = blockIdx.x * 32;       // 128 N-tiles of 32 cols covering 4096
  int ks0  = blockIdx.y * KPER;     // K chunk
  int lo   = lane & 15;
  int kh   = (lane >> 4) << 1;

  const float* wp = w1 + ((size_t)(ks0 * 4) + kh) * 4096 + nb + lo;
  const float* sp = s + (size_t)lo * PP + (size_t)(ks0 * 4) + kh;

  v8f acc00 = {}, acc01 = {}, acc10 = {}, acc11 = {};

#pragma unroll 1
  for (int blk = 0; blk < KPER / TBLK; ++blk) {
    v2f b0[TBLK], b1[TBLK], a0[TBLK], a1[TBLK];
#pragma unroll
    for (int t = 0; t < TBLK; ++t) {            // load phase: 48 loads
      const float* wpt = wp + (size_t)t * (4 * 4096);
      const float* spt = sp + t * 4;
      b0[t].x = __builtin_nontemporal_load(wpt);
      b1[t].x = __builtin_nontemporal_load(wpt + 16);
      b0[t].y = __builtin_nontemporal_load(wpt + 4096);
      b1[t].y = __builtin_nontemporal_load(wpt + 4096 + 16);
      a0[t].x = spt[0];          a0[t].y = spt[1];
      a1[t].x = spt[16 * PP];    a1[t].y = spt[16 * PP + 1];
    }
#pragma unroll
    for (int t = 0; t < TBLK; ++t) {            // compute phase: 32 WMMAs
      acc00 = wmma_f32(a0[t], b0[t], acc00);
      acc01 = wmma_f32(a0[t], b1[t], acc01);
      acc10 = wmma_f32(a1[t], b0[t], acc10);
      acc11 = wmma_f32(a1[t], b1[t], acc11);
    }
    wp += (size_t)TBLK * (4 * 4096);
    sp += TBLK * 4;
  }
  // tail: KPER % TBLK = 7 k-steps
#pragma unroll
  for (int t = 0; t < KPER % TBLK; ++t) {
    v2f b0, b1, a0, a1;
    const float* wpt = wp + (size_t)t * (4 * 4096);
    const float* spt = sp + t * 4;
    b0.x = __builtin_nontemporal_load(wpt);
    b1.x = __builtin_nontemporal_load(wpt + 16);
    b0.y = __builtin_nontemporal_load(wpt + 4096);
    b1.y = __builtin_nontemporal_load(wpt + 4096 + 16);
    a0.x = spt[0];          a0.y = spt[1];
    a1.x = spt[16 * PP];    a1.y = spt[16 * PP + 1];
    acc00 = wmma_f32(a0, b0, acc00);
    acc01 = wmma_f32(a0, b1, acc01);
    acc10 = wmma_f32(a1, b0, acc10);
    acc11 = wmma_f32(a1, b1, acc11);
  }

  int roff = (lane >> 4) * 8;
#pragma unroll
  for (int v = 0; v < 8; ++v) {
    atomicAdd(&c1[(size_t)(roff + v) * 4096 + nb + lo],           acc00[v]);
    atomicAdd(&c1[(size_t)(roff + v) * 4096 + nb + 16 + lo],      acc01[v]);
    atomicAdd(&c1[(size_t)(16 + roff + v) * 4096 + nb + lo],      acc10[v]);
    atomicAdd(&c1[(size_t)(16 + roff + v) * 4096 + nb + 16 + lo], acc11[v]);
  }
}

// ---------------- kernel 4: bias + rrelu on c1 ----------------
__global__ void act1_kernel(float* __restrict__ c1,
                            const float* __restrict__ b1) {
  int idx = blockIdx.x * blockDim.x + threadIdx.x;
  if (idx < BB * 4096) c1[idx] = rrelu(c1[idx] + b1[idx & 4095]);
}

// ---------------- kernel 5: c2 = relu(c1[32,4096] @ w2[4096,256] + b2) -----
__global__ __launch_bounds__(32) void cls2_kernel(
    const float* __restrict__ c1, const float* __restrict__ w2,
    const float* __restrict__ b2, float* __restrict__ c2) {
  int lane = threadIdx.x;
  int nb   = blockIdx.x * 32;       // 8 tiles of 32 cols covering 256
  int lo   = lane & 15, kh = (lane >> 4) << 1;

  const float* wp = w2 + kh * 256 + nb + lo;
  const float* cp = c1 + lo * 4096 + kh;

  v8f acc00 = {}, acc01 = {}, acc10 = {}, acc11 = {};
#pragma unroll 1
  for (int blk = 0; blk < 1024 / TBLK; ++blk) {
    v2f b0[TBLK], b1[TBLK], a0[TBLK], a1[TBLK];
#pragma unroll
    for (int t = 0; t < TBLK; ++t) {
      const float* wpt = wp + t * (4 * 256);
      const float* cpt = cp + t * 4;
      b0[t].x = wpt[0];    b1[t].x = wpt[16];
      b0[t].y = wpt[256];  b1[t].y = wpt[256 + 16];
      a0[t].x = cpt[0];          a0[t].y = cpt[1];
      a1[t].x = cpt[16 * 4096];  a1[t].y = cpt[16 * 4096 + 1];
    }
#pragma unroll
    for (int t = 0; t < TBLK; ++t) {
      acc00 = wmma_f32(a0[t], b0[t], acc00);
      acc01 = wmma_f32(a0[t], b1[t], acc01);
      acc10 = wmma_f32(a1[t], b0[t], acc10);
      acc11 = wmma_f32(a1[t], b1[t], acc11);
    }
    wp += TBLK * (4 * 256);
    cp += TBLK * 4;
  }
  int roff = (lane >> 4) * 8;
  float bias0 = b2[nb + lo], bias1 = b2[nb + 16 + lo];
#pragma unroll
  for (int v = 0; v < 8; ++v) {
    c2[(roff + v) * 256 + nb + lo]           = fmaxf(acc00[v] + bias0, 0.0f);
    c2[(roff + v) * 256 + nb + 16 + lo]      = fmaxf(acc01[v] + bias1, 0.0f);
    c2[(16 + roff + v) * 256 + nb + lo]      = fmaxf(acc10[v] + bias0, 0.0f);
    c2[(16 + roff + v) * 256 + nb + 16 + lo] = fmaxf(acc11[v] + bias1, 0.0f);
  }
}

// ---------------- kernel 6: c3 = relu(c2[32,256] @ w3[256,64] + b3) --------
__global__ __launch_bounds__(32) void cls3_kernel(
    const float* __restrict__ c2, const float* __restrict__ w3,
    const float* __restrict__ b3, float* __restrict__ c3) {
  int lane = threadIdx.x;
  int nb   = blockIdx.x * 16;       // 4 N-tiles covering 64
  int lo   = lane & 15, kh = (lane >> 4) << 1;

  const float* wp = w3 + kh * 64 + nb + lo;
  const float* cp = c2 + lo * 256 + kh;

  v8f acc0 = {}, acc1 = {};
#pragma unroll 4
  for (int ks = 0; ks < 64; ++ks) {
    v2f bf, a0, a1;
    bf.x = wp[0];  bf.y = wp[64];
    a0.x = cp[0];         a0.y = cp[1];
    a1.x = cp[16 * 256];  a1.y = cp[16 * 256 + 1];
    acc0 = wmma_f32(a0, bf, acc0);
    acc1 = wmma_f32(a1, bf, acc1);
    wp += 256;
    cp += 4;
  }
  int roff = (lane >> 4) * 8;
  float bias = b3[nb + lo];
#pragma unroll
  for (int v = 0; v < 8; ++v) {
    c3[(roff + v) * 64 + nb + lo]      = fmaxf(acc0[v] + bias, 0.0f);
    c3[(16 + roff + v) * 64 + nb + lo] = fmaxf(acc1[v] + bias, 0.0f);
  }
}

// -------- kernel 7: final 64->3 + log_softmax, one subject per lane --------
__global__ __launch_bounds__(32) void cls4_kernel(
    const float* __restrict__ c3, const float* __restrict__ w4,
    const float* __restrict__ b4, float* __restrict__ out) {
  int b = threadIdx.x;   // 32 subjects, one wave
  float o[3];
  for (int k = 0; k < 3; ++k) {
    float acc = b4[k];
    for (int j = 0; j < 64; ++j) acc += c3[b * 64 + j] * w4[j * 3 + k];
    o[k] = acc;
  }
  float m   = fmaxf(o[0], fmaxf(o[1], o[2]));
  float lse = logf(expf(o[0] - m) + expf(o[1] - m) + expf(o[2] - m));
  for (int k = 0; k < 3; ++k) out[b * 3 + k] = o[k] - m - lse;
}

// ---------------- launch ----------------
extern "C" void kernel_launch(void* const* d_in, const int* in_sizes, int n_in,
                              void* d_out, int out_size, void* d_ws,
                              size_t ws_size, hipStream_t stream) {
  const float* x     = (const float*)d_in[0];
  const float* fe_w1 = (const float*)d_in[1];
  const float* fe_b1 = (const float*)d_in[2];
  const float* fe_w2 = (const float*)d_in[3];
  const float* fe_b2 = (const float*)d_in[4];
  const float* fe_w3 = (const float*)d_in[5];
  const float* fe_b3 = (const float*)d_in[6];
  const float* sm_w1 = (const float*)d_in[7];
  const float* sm_b1 = (const float*)d_in[8];
  const float* sm_w2 = (const float*)d_in[9];
  const float* sm_b2 = (const float*)d_in[10];
  const float* sm_w3 = (const float*)d_in[11];
  const float* sm_b3 = (const float*)d_in[12];
  const float* sm_w4 = (const float*)d_in[13];
  const float* sm_b4 = (const float*)d_in[14];
  const float* cl_w1 = (const float*)d_in[15];
  const float* cl_b1 = (const float*)d_in[16];
  const float* cl_w2 = (const float*)d_in[17];
  const float* cl_b2 = (const float*)d_in[18];
  const float* cl_w3 = (const float*)d_in[19];
  const float* cl_b3 = (const float*)d_in[20];
  const float* cl_w4 = (const float*)d_in[21];
  const float* cl_b4 = (const float*)d_in[22];

  char* w = (char*)d_ws;
  float* feats  = (float*)w; w += (size_t)ROWS * 32 * sizeof(float);
  float* s      = (float*)w; w += (size_t)BB * PP * sizeof(float);
  float* c1     = (float*)w; w += (size_t)BB * 4096 * sizeof(float);
  float* c2     = (float*)w; w += (size_t)BB * 256 * sizeof(float);
  float* c3     = (float*)w; w += (size_t)BB * 64 * sizeof(float);
  float* fe_w1p = (float*)w; w += (size_t)DDP * 64 * sizeof(float);
  float* sm_w3p = (float*)w; w += (size_t)16 * 16 * sizeof(float);
  int*   pi1    = (int*)w;   w += (size_t)PP * sizeof(int);
  int*   pi2    = (int*)w;   w += (size_t)PP * sizeof(int);

  prep_kernel<<<(BB * 4096 + 255) / 256, 256, 0, stream>>>(
      pi1, pi2, c1, fe_w1, fe_w1p, sm_w3, sm_w3p);

  fe_kernel<<<ROWS / 16, 32, 0, stream>>>(x, fe_w1p, fe_b1, fe_w2, fe_b2,
                                          fe_w3, fe_b3, feats);

  sim_kernel<<<dim3((PP + 15) / 16, BB), 32, 0, stream>>>(
      feats, pi1, pi2, sm_w1, sm_b1, sm_w2, sm_b2, sm_w3p, sm_b3, sm_w4, sm_b4,
      s);

  cls1_kernel<<<dim3(4096 / 32, KCHUNKS), 32, 0, stream>>>(s, cl_w1, c1);
  act1_kernel<<<(BB * 4096 + 255) / 256, 256, 0, stream>>>(c1, cl_b1);
  cls2_kernel<<<256 / 32, 32, 0, stream>>>(c1, cl_w2, cl_b2, c2);
  cls3_kernel<<<64 / 16, 32, 0, stream>>>(c2, cl_w3, cl_b3, c3);
  cls4_kernel<<<1, 32, 0, stream>>>(c3, cl_w4, cl_b4, (float*)d_out);
}